// GraphConv_12120397709963
// MI455X (gfx1250) — compile-verified
//
#include <hip/hip_runtime.h>

typedef float v2f __attribute__((ext_vector_type(2)));
typedef float v8f __attribute__((ext_vector_type(8)));

#define N_NODES 150000
#define N_EDGES 1200000
#define D 64

// ---------------------------------------------------------------------------
// Kernel 1: zero the aggregation buffer (we aggregate directly into d_out,
// which is exactly N_NODES * D floats, so no scratch dependence).
// ---------------------------------------------------------------------------
__global__ __launch_bounds__(256) void zero_kernel(float4* __restrict__ out, int n4) {
    int i = blockIdx.x * blockDim.x + threadIdx.x;
    if (i < n4) out[i] = make_float4(0.f, 0.f, 0.f, 0.f);
}

// ---------------------------------------------------------------------------
// Kernel 2: SpMM scatter. 16 threads per edge, one float4 chunk each.
// x (38.4 MB) and the aggregate (38.4 MB) are both L2-resident (192 MB L2),
// so the gathers and the fp32 atomic adds resolve in L2's atomic units.
// ---------------------------------------------------------------------------
__global__ __launch_bounds__(256) void spmm_scatter(const float* __restrict__ x,
                                                    const int*   __restrict__ row,
                                                    const int*   __restrict__ col,
                                                    const float* __restrict__ vals,
                                                    float*       __restrict__ agg,
                                                    int nE) {
    int t = blockIdx.x * blockDim.x + threadIdx.x;
    int e = t >> 4;
    if (e >= nE) return;
    int chunk = (t & 15) << 2;                 // 0,4,8,...,60
    int c = col[e];
    int r = row[e];
    float v = vals[e];
    const float4 xv = *reinterpret_cast<const float4*>(x + (size_t)c * D + chunk);
    float* dst = agg + (size_t)r * D + chunk;
    unsafeAtomicAdd(dst + 0, v * xv.x);        // -> global_atomic_add_f32
    unsafeAtomicAdd(dst + 1, v * xv.y);
    unsafeAtomicAdd(dst + 2, v * xv.z);
    unsafeAtomicAdd(dst + 3, v * xv.w);
}

// ---------------------------------------------------------------------------
// Kernel 3: in-place projection out = agg @ W^T + b using fp32 WMMA.
// One wave per 16 rows. All A fragments (16 rows x 64 cols) are loaded into
// registers BEFORE any store, so the in-place transform is race-free (waves
// own disjoint 16-row blocks).
//
// V_WMMA_F32_16X16X4_F32 layouts (ISA 7.12.2, 32-bit):
//   A (16x4): lanes 0-15 -> M=lane, VGPR0=K0, VGPR1=K1
//             lanes 16-31 -> M=lane-16, VGPR0=K2, VGPR1=K3
//   B (4x16): mirrored with N in place of M  (B[k][n] = W[n][k])
//   C/D (16x16): VGPR j: lanes 0-15 -> M=j, lanes 16-31 -> M=j+8, N=lane&15
// ---------------------------------------------------------------------------
__global__ __launch_bounds__(256) void gemm_wmma(float* __restrict__ out,
                                                 const float* __restrict__ W,
                                                 const float* __restrict__ b,
                                                 int nRows) {
    const int lane = threadIdx.x & 31;
    const int wave = (blockIdx.x * blockDim.x + threadIdx.x) >> 5;
    const int m0 = wave * 16;
    if (m0 >= nRows) return;                   // whole wave exits -> EXEC stays all-1s

    const int lr   = lane & 15;                // M (or N) index within half-wave
    const int koff = (lane >> 4) * 2;          // K sub-offset: 0 or 2

    // ---- load all A fragments first (in-place safety) ----
    v2f a[16];
    const float* arow = out + (size_t)(m0 + lr) * D;
    #pragma unroll
    for (int ks = 0; ks < 16; ++ks) {
        const float2 t = *reinterpret_cast<const float2*>(arow + ks * 4 + koff);
        a[ks][0] = t.x;
        a[ks][1] = t.y;
    }

    // ---- four 16-wide N tiles ----
    #pragma unroll
    for (int nt = 0; nt < 4; ++nt) {
        const int n0 = nt * 16;
        const float bias = b[n0 + lr];
        v8f acc = {bias, bias, bias, bias, bias, bias, bias, bias};

        const float* wrow = W + (size_t)(n0 + lr) * D;   // B[k][n] = W[n][k]
        #pragma unroll
        for (int ks = 0; ks < 16; ++ks) {
            const float2 t = *reinterpret_cast<const float2*>(wrow + ks * 4 + koff);
            v2f bf;
            bf[0] = t.x;
            bf[1] = t.y;
            // 8 args: (neg_a, A, neg_b, B, c_mod, C, reuse_a, reuse_b)
            acc = __builtin_amdgcn_wmma_f32_16x16x4_f32(false, a[ks], false, bf,
                                                        (short)0, acc, false, false);
        }

        // ---- store the 16x16 tile ----
        const int mhi = (lane >> 4) << 3;      // +8 rows for upper half-wave
        #pragma unroll
        for (int j = 0; j < 8; ++j) {
            out[(size_t)(m0 + j + mhi) * D + n0 + lr] = acc[j];
        }
    }
}

extern "C" void kernel_launch(void* const* d_in, const int* in_sizes, int n_in,
                              void* d_out, int out_size, void* d_ws, size_t ws_size,
                              hipStream_t stream) {
    const float* x    = (const float*)d_in[0];
    const int*   row  = (const int*)  d_in[1];
    const int*   col  = (const int*)  d_in[2];
    const float* vals = (const float*)d_in[3];
    const float* W    = (const float*)d_in[4];
    const float* b    = (const float*)d_in[5];
    float* out = (float*)d_out;

    // 1) zero the aggregate (held in d_out)
    const int n4 = N_NODES * D / 4;            // 2,400,000 float4s
    zero_kernel<<<(n4 + 255) / 256, 256, 0, stream>>>((float4*)out, n4);

    // 2) scatter-add SpMM into d_out
    const int spmm_threads = N_EDGES * 16;     // 19,200,000
    spmm_scatter<<<(spmm_threads + 255) / 256, 256, 0, stream>>>(x, row, col, vals,
                                                                 out, N_EDGES);

    // 3) in-place WMMA projection: out = out @ W^T + b
    const int waves = N_NODES / 16;            // 9,375 waves
    gemm_wmma<<<(waves * 32 + 255) / 256, 256, 0, stream>>>(out, W, b, N_NODES);
}